// ActivationGatedGCNLayer_83476984365535
// MI455X (gfx1250) — compile-verified
//
#include <hip/hip_runtime.h>
#include <stdint.h>

#define DIM 32
#define BN_EPS 1e-5f

typedef float v4f __attribute__((ext_vector_type(4)));

// stats layout in workspace (floats):
//   [  0: 32) sum_e    [ 32: 64) sumsq_e
//   [ 64: 96) sum_h    [ 96:128) sumsq_h
//   [128:160) scale_h  [160:192) bias_h
//   [192:224) scale_e  [224:256) bias_e
//
// agg layout (interleaved per node, 64 floats = 256 B per node):
//   agg[n*64 + 0  + c] = num[n][c]
//   agg[n*64 + 32 + c] = den[n][c]

// ---------------------------------------------------------------------------
// CDNA5 async-copy helpers (gfx1250): global -> LDS DMA tracked by ASYNCcnt.
// Non-temporal hint: this path streams data we will touch exactly once.
// ---------------------------------------------------------------------------
__device__ __forceinline__ void async_load_b128_nt(uint32_t lds_byte_addr, const void* gaddr) {
    // vdst = VGPR holding LDS byte address, vaddr = 64-bit global address
    asm volatile("global_load_async_to_lds_b128 %0, %1, off th:TH_LOAD_NT"
                 :: "v"(lds_byte_addr), "v"(gaddr)
                 : "memory");
}
__device__ __forceinline__ void wait_asynccnt_le1() {
    asm volatile("s_wait_asynccnt 0x1" ::: "memory");
}
__device__ __forceinline__ void wait_asynccnt_0() {
    asm volatile("s_wait_asynccnt 0x0" ::: "memory");
}

// ---------------------------------------------------------------------------
// Edge pass: one wave32 per edge, lane = feature column.
//   e_ij = e + h*norm[src] + h*norm[dst]   (NT-written pre-BN to e region)
//   sigma = sigmoid(e_ij)
//   agg[dst].num += sigma * h*norm[src] ; agg[dst].den += sigma  (L2 atomics)
//   per-lane running sum / sumsq of e_ij for BatchNorm stats
// ---------------------------------------------------------------------------
__global__ void gcn_edge_kernel(const float* __restrict__ h,
                                const float* __restrict__ e,
                                const float* __restrict__ norm,
                                const int*   __restrict__ src,
                                const int*   __restrict__ dst,
                                float* __restrict__ e_pre,
                                float* __restrict__ agg,
                                float* __restrict__ stats,
                                int n_edges)
{
    const int lane = threadIdx.x & 31;
    const int wid  = (int)((blockIdx.x * blockDim.x + threadIdx.x) >> 5);
    const int nw   = (int)((gridDim.x * blockDim.x) >> 5);

    float s1 = 0.0f, s2 = 0.0f;
    for (int idx = wid; idx < n_edges; idx += nw) {
        const int s = src[idx];
        const int d = dst[idx];
        // prefetch the gathered rows of the next iteration (global_prefetch_b8)
        const int nxt = idx + nw;
        if (nxt < n_edges) {
            __builtin_prefetch(&h[(size_t)src[nxt] * DIM], 0, 0);
            __builtin_prefetch(&h[(size_t)dst[nxt] * DIM], 0, 0);
        }
        const float hs  = h[(size_t)s * DIM + lane] * norm[s];
        const float hd  = h[(size_t)d * DIM + lane] * norm[d];
        // e is streamed exactly once: non-temporal so it never thrashes L2
        const float eij = __builtin_nontemporal_load(&e[(size_t)idx * DIM + lane]) + hs + hd;
        __builtin_nontemporal_store(eij, &e_pre[(size_t)idx * DIM + lane]);

        const float sig = 1.0f / (1.0f + __expf(-eij));
        atomicAdd(&agg[(size_t)d * 64 + lane],      sig * hs);
        atomicAdd(&agg[(size_t)d * 64 + 32 + lane], sig);

        s1 += eij;
        s2 += eij * eij;
    }
    atomicAdd(&stats[lane],       s1);
    atomicAdd(&stats[DIM + lane], s2);
}

// ---------------------------------------------------------------------------
// Node pass: h_new = (h*norm + num/(den+1e-6)) * norm ; accumulate h stats.
// ---------------------------------------------------------------------------
__global__ void gcn_node_kernel(const float* __restrict__ h,
                                const float* __restrict__ norm,
                                const float* __restrict__ agg,
                                float* __restrict__ h_pre,
                                float* __restrict__ stats,
                                int n_nodes)
{
    const int lane = threadIdx.x & 31;
    const int wid  = (int)((blockIdx.x * blockDim.x + threadIdx.x) >> 5);
    const int nw   = (int)((gridDim.x * blockDim.x) >> 5);

    float s1 = 0.0f, s2 = 0.0f;
    for (int n = wid; n < n_nodes; n += nw) {
        const float nm = norm[n];
        const float v  = h[(size_t)n * DIM + lane] * nm;
        const float a  = agg[(size_t)n * 64 + lane];
        const float b  = agg[(size_t)n * 64 + 32 + lane];
        const float hn = (v + a / (b + 1e-6f)) * nm;
        h_pre[(size_t)n * DIM + lane] = hn;
        s1 += hn;
        s2 += hn * hn;
    }
    atomicAdd(&stats[64 + lane], s1);
    atomicAdd(&stats[96 + lane], s2);
}

// ---------------------------------------------------------------------------
// Fold batch statistics into per-column affine: y = x*scale + bias.
// ---------------------------------------------------------------------------
__global__ void gcn_stats_kernel(const float* __restrict__ gamma_h,
                                 const float* __restrict__ beta_h,
                                 const float* __restrict__ gamma_e,
                                 const float* __restrict__ beta_e,
                                 float* __restrict__ stats,
                                 int n_nodes, int n_edges)
{
    const int t = threadIdx.x;
    if (t < DIM) {
        const float inv  = 1.0f / (float)n_edges;
        const float mean = stats[t] * inv;
        const float var  = stats[DIM + t] * inv - mean * mean;
        const float sc   = gamma_e[t] * rsqrtf(var + BN_EPS);
        stats[192 + t] = sc;
        stats[224 + t] = beta_e[t] - mean * sc;
    } else if (t < 2 * DIM) {
        const int c = t - DIM;
        const float inv  = 1.0f / (float)n_nodes;
        const float mean = stats[64 + c] * inv;
        const float var  = stats[96 + c] * inv - mean * mean;
        const float sc   = gamma_h[c] * rsqrtf(var + BN_EPS);
        stats[128 + c] = sc;
        stats[160 + c] = beta_h[c] - mean * sc;
    }
}

// ---------------------------------------------------------------------------
// BN+ReLU in place for the (small) node tensor: plain float4 stream.
// stride % 8 == 0 so each thread's column phase (i & 7) is loop-invariant.
// ---------------------------------------------------------------------------
__global__ void gcn_bn_h_kernel(float* __restrict__ x,
                                const float* __restrict__ stats,
                                size_t n4)
{
    const size_t tid    = (size_t)blockIdx.x * blockDim.x + threadIdx.x;
    const size_t stride = (size_t)gridDim.x * blockDim.x;
    const int cb = (int)(tid & 7) * 4;
    const float sc0 = stats[128 + cb + 0], sc1 = stats[128 + cb + 1];
    const float sc2 = stats[128 + cb + 2], sc3 = stats[128 + cb + 3];
    const float bi0 = stats[160 + cb + 0], bi1 = stats[160 + cb + 1];
    const float bi2 = stats[160 + cb + 2], bi3 = stats[160 + cb + 3];
    for (size_t i = tid; i < n4; i += stride) {
        v4f v = ((const v4f*)x)[i];
        v.x = fmaxf(fmaf(v.x, sc0, bi0), 0.0f);
        v.y = fmaxf(fmaf(v.y, sc1, bi1), 0.0f);
        v.z = fmaxf(fmaf(v.z, sc2, bi2), 0.0f);
        v.w = fmaxf(fmaf(v.w, sc3, bi3), 0.0f);
        ((v4f*)x)[i] = v;
    }
}

// ---------------------------------------------------------------------------
// BN+ReLU in place for the (huge) edge tensor: double-buffered async-to-LDS
// pipeline. Each wave owns a 32-float4 (512 B) tile per stage; the async DMA
// for tile k+1 overlaps the LDS-read/transform/store of tile k. All traffic
// non-temporal: this stream is touched exactly once here.
// ---------------------------------------------------------------------------
__global__ void gcn_bn_e_kernel(float* __restrict__ x,
                                const float* __restrict__ stats,
                                size_t n4)
{
    __shared__ v4f sbuf[2][8][32];      // [buffer][wave][lane] = 8 KB

    const int lane = threadIdx.x & 31;
    const int wave = threadIdx.x >> 5;
    const size_t wid = ((size_t)blockIdx.x * blockDim.x + threadIdx.x) >> 5;
    const size_t nw  = ((size_t)gridDim.x * blockDim.x) >> 5;
    const size_t ntiles = (n4 + 31) >> 5;

    // nw*32 is a multiple of 8 -> this lane's columns are loop-invariant
    const int cb = (lane & 7) * 4;
    const float sc0 = stats[192 + cb + 0], sc1 = stats[192 + cb + 1];
    const float sc2 = stats[192 + cb + 2], sc3 = stats[192 + cb + 3];
    const float bi0 = stats[224 + cb + 0], bi1 = stats[224 + cb + 1];
    const float bi2 = stats[224 + cb + 2], bi3 = stats[224 + cb + 3];

    // LDS byte addresses for this lane's two staging slots (flat addr low 32
    // bits == workgroup-relative LDS offset on CDNA5)
    uint32_t laddr0 = (uint32_t)(uintptr_t)&sbuf[0][wave][lane];
    uint32_t laddr1 = (uint32_t)(uintptr_t)&sbuf[1][wave][lane];

    if (wid >= ntiles) return;          // wave-uniform exit

    // prologue: stage first tile (clamp tail lanes into range; stores guarded)
    {
        size_t i0 = wid * 32 + (size_t)lane;
        if (i0 >= n4) i0 = n4 - 1;
        async_load_b128_nt(laddr0, (const void*)((const v4f*)x + i0));
    }

    int cur = 0;
    for (size_t t = wid; t < ntiles; t += nw) {
        const size_t tn = t + nw;
        const bool more = (tn < ntiles);  // wave-uniform
        if (more) {
            size_t in = tn * 32 + (size_t)lane;
            if (in >= n4) in = n4 - 1;
            async_load_b128_nt(cur ? laddr0 : laddr1,
                               (const void*)((const v4f*)x + in));
            wait_asynccnt_le1();          // oldest DMA (current tile) done
        } else {
            wait_asynccnt_0();
        }

        v4f v = sbuf[cur][wave][lane];    // ds_load_b128 from staged tile
        v.x = fmaxf(fmaf(v.x, sc0, bi0), 0.0f);
        v.y = fmaxf(fmaf(v.y, sc1, bi1), 0.0f);
        v.z = fmaxf(fmaf(v.z, sc2, bi2), 0.0f);
        v.w = fmaxf(fmaf(v.w, sc3, bi3), 0.0f);

        const size_t i = t * 32 + (size_t)lane;
        if (i < n4) __builtin_nontemporal_store(v, (v4f*)x + i);

        cur ^= 1;
    }
}

// ---------------------------------------------------------------------------
// Host-side launch (graph-capture safe: only async ops on `stream`).
// ---------------------------------------------------------------------------
extern "C" void kernel_launch(void* const* d_in, const int* in_sizes, int n_in,
                              void* d_out, int out_size, void* d_ws, size_t ws_size,
                              hipStream_t stream)
{
    const float* h       = (const float*)d_in[0];
    const float* e       = (const float*)d_in[1];
    const float* norm    = (const float*)d_in[2];
    const int*   src     = (const int*)d_in[3];
    const int*   dst     = (const int*)d_in[4];
    const float* gamma_h = (const float*)d_in[5];
    const float* beta_h  = (const float*)d_in[6];
    const float* gamma_e = (const float*)d_in[7];
    const float* beta_e  = (const float*)d_in[8];

    const int n_nodes = in_sizes[0] / DIM;
    const int n_edges = in_sizes[1] / DIM;

    // output layout: h_out first, then e_out (flat, return order)
    float* h_pre = (float*)d_out;
    float* e_pre = h_pre + (size_t)n_nodes * DIM;

    // workspace: agg (num/den interleaved, 64 floats per node) | stats(256)
    float* agg   = (float*)d_ws;
    float* stats = agg + (size_t)2 * n_nodes * DIM;

    const size_t zero_bytes = ((size_t)2 * n_nodes * DIM + 256) * sizeof(float);
    (void)hipMemsetAsync(d_ws, 0, zero_bytes, stream);

    gcn_edge_kernel<<<2048, 256, 0, stream>>>(h, e, norm, src, dst,
                                              e_pre, agg, stats, n_edges);
    gcn_node_kernel<<<512, 256, 0, stream>>>(h, norm, agg,
                                             h_pre, stats, n_nodes);
    gcn_stats_kernel<<<1, 64, 0, stream>>>(gamma_h, beta_h, gamma_e, beta_e,
                                           stats, n_nodes, n_edges);

    const size_t n4h = (size_t)n_nodes * DIM / 4;
    gcn_bn_h_kernel<<<512, 256, 0, stream>>>(h_pre, stats, n4h);

    const size_t n4e = (size_t)n_edges * DIM / 4;
    gcn_bn_e_kernel<<<1024, 256, 0, stream>>>(e_pre, stats, n4e);
}